// observation_processing_network_68813966017023
// MI455X (gfx1250) — compile-verified
//
#include <hip/hip_runtime.h>

#define HD 64     // H*D
#define NHEAD 8
#define DPH 8
#define NEG_SLOPE 0.2f
#define APAD 4    // LDS row padding (dwords) to spread banks

typedef __attribute__((ext_vector_type(2))) float v2f;
typedef __attribute__((ext_vector_type(8))) float v8f;

// ---------- monotone float<->uint encoding for atomicMax on f32 ----------
__device__ __forceinline__ unsigned f2ord(float f) {
  unsigned u = __float_as_uint(f);
  return (u & 0x80000000u) ? ~u : (u | 0x80000000u);
}
__device__ __forceinline__ float ord2f(unsigned u) {
  return __uint_as_float((u & 0x80000000u) ? (u & 0x7fffffffu) : ~u);
}

// ---------- GEMM: O[M,64] = X[M,K] @ W[K,64] ----------
// One wave per 16-row strip; 4 column tiles (16x16) per wave sharing one A
// fragment -> 4 v_wmma_f32_16x16x4_f32 per K-step. A strip staged via LDS
// (contiguous coalesced global read, bank-padded LDS rows).
__global__ __launch_bounds__(128) void gat_gemm_wmma(const float* __restrict__ X,
                                                     const float* __restrict__ W,
                                                     float* __restrict__ O,
                                                     int M, int K, int kshift) {
#ifdef __HIP_DEVICE_COMPILE__
  __shared__ float smem[4 * 16 * (128 + APAD)];
  const int wid   = threadIdx.x >> 5;
  const int lane  = threadIdx.x & 31;
  const int strip = blockIdx.x * 4 + wid;
  if (strip * 16 >= M) return;                 // wave-uniform; EXEC stays all-1s
  const int rstride = K + APAD;
  float* tile = smem + wid * 16 * (128 + APAD);

  // --- stage 16xK A-strip: contiguous in global, padded rows in LDS ---
  const float* gsrc = X + (size_t)strip * 16 * K;
  const int nf4 = (16 * K) >> 2;               // float4 chunks in strip
  for (int f = lane; f < nf4; f += 32) {
    const int el = f << 2;
    const int r  = el >> kshift;               // el / K
    const int c  = el & (K - 1);               // el % K
    const float4 v = *(const float4*)(gsrc + el);
    float* dp = tile + r * rstride + c;
    dp[0] = v.x; dp[1] = v.y; dp[2] = v.z; dp[3] = v.w;
  }

  // --- compute: A 16x4 frag from LDS, B 4x16 frags from global (W cached) ---
  const int hl = lane >> 4;                    // half-wave
  const int l  = lane & 15;
  v8f acc0 = {0.f,0.f,0.f,0.f,0.f,0.f,0.f,0.f};
  v8f acc1 = acc0, acc2 = acc0, acc3 = acc0;
  for (int k0 = 0; k0 < K; k0 += 4) {
    // A layout: lanes0-15 M=0..15 {K=k0,k0+1}; lanes16-31 {K=k0+2,k0+3}
    const float2 af = *(const float2*)(tile + l * rstride + k0 + hl * 2);
    v2f a; a.x = af.x; a.y = af.y;
    // B layout: VGPR0 row k0+2*hl, VGPR1 row k0+2*hl+1; N = lane&15
    const float* wrow = W + (size_t)(k0 + hl * 2) * HD + l;
    v2f b0, b1, b2, b3;
    b0.x = wrow[0];       b0.y = wrow[HD];
    b1.x = wrow[16];      b1.y = wrow[16 + HD];
    b2.x = wrow[32];      b2.y = wrow[32 + HD];
    b3.x = wrow[48];      b3.y = wrow[48 + HD];
    acc0 = __builtin_amdgcn_wmma_f32_16x16x4_f32(false, a, false, b0, (short)0, acc0, false, false);
    acc1 = __builtin_amdgcn_wmma_f32_16x16x4_f32(false, a, false, b1, (short)0, acc1, false, false);
    acc2 = __builtin_amdgcn_wmma_f32_16x16x4_f32(false, a, false, b2, (short)0, acc2, false, false);
    acc3 = __builtin_amdgcn_wmma_f32_16x16x4_f32(false, a, false, b3, (short)0, acc3, false, false);
  }

  // C/D layout: VGPR r -> row strip*16 + hl*8 + r, col tile + (lane&15)
  float* obase = O + (size_t)(strip * 16 + hl * 8) * HD + l;
  #pragma unroll
  for (int r = 0; r < 8; ++r) {
    obase[(size_t)r * HD + 0]  = acc0[r];
    obase[(size_t)r * HD + 16] = acc1[r];
    obase[(size_t)r * HD + 32] = acc2[r];
    obase[(size_t)r * HD + 48] = acc3[r];
  }
#endif
}

// ---------- per-node attention logits: es/ed[n,h] = <h[n,h,:], att[0/1,h,:]> ----------
__global__ void gat_node_logits(const float* __restrict__ hfeat, const float* __restrict__ att,
                                float* __restrict__ es, float* __restrict__ ed, int N8) {
  int i = blockIdx.x * blockDim.x + threadIdx.x;
  if (i >= N8) return;
  const int n = i >> 3, h = i & 7;
  const float* hp = hfeat + (size_t)n * HD + h * DPH;
  const float* a0 = att + h * DPH;        // att[0]
  const float* a1 = att + HD + h * DPH;   // att[1]
  float s0 = 0.f, s1 = 0.f;
  #pragma unroll
  for (int j = 0; j < DPH; ++j) { s0 += hp[j] * a0[j]; s1 += hp[j] * a1[j]; }
  es[i] = s0;
  ed[i] = s1;
}

// ---------- init: m = enc(-inf), s = 0, agg = 0 ----------
__global__ void gat_init(float* __restrict__ ssum, unsigned* __restrict__ mx,
                         float* __restrict__ agg, int N8, int N64) {
  int i = blockIdx.x * blockDim.x + threadIdx.x;
  if (i < N8) { ssum[i] = 0.f; mx[i] = 0x007FFFFFu; }  // enc(-inf)
  if (i < N64) agg[i] = 0.f;
}

// ---------- edge pass 1: logits + leaky relu + segment max ----------
__global__ void gat_edge_max(const int* __restrict__ src, const int* __restrict__ dst,
                             const float* __restrict__ es, const float* __restrict__ ed,
                             float* __restrict__ ew, unsigned* __restrict__ mx, int EH) {
  int i = blockIdx.x * blockDim.x + threadIdx.x;
  if (i >= EH) return;
  const int e = i >> 3, h = i & 7;
  const int s = src[e], d = dst[e];
  float v = es[s * NHEAD + h] + ed[d * NHEAD + h];
  v = (v > 0.f) ? v : NEG_SLOPE * v;
  ew[i] = v;
  atomicMax(&mx[d * NHEAD + h], f2ord(v));
}

// ---------- edge pass 2: w = exp(e - m[dst]); segment sum ----------
__global__ void gat_edge_exp(const int* __restrict__ dst, float* __restrict__ ew,
                             const unsigned* __restrict__ mx, float* __restrict__ ssum, int EH) {
  int i = blockIdx.x * blockDim.x + threadIdx.x;
  if (i >= EH) return;
  const int e = i >> 3, h = i & 7;
  const int d = dst[e];
  const float w = __expf(ew[i] - ord2f(mx[d * NHEAD + h]));
  ew[i] = w;
  atomicAdd(&ssum[d * NHEAD + h], w);
}

// ---------- edge pass 3: alpha = w/(s+eps); out[dst] += alpha * h[src] ----------
__global__ void gat_edge_scatter(const int* __restrict__ src, const int* __restrict__ dst,
                                 const float* __restrict__ ew, const float* __restrict__ ssum,
                                 const float* __restrict__ hfeat, float* __restrict__ out, int EH) {
  int i = blockIdx.x * blockDim.x + threadIdx.x;
  if (i >= EH) return;
  const int e = i >> 3, h = i & 7;
  const int s = src[e], d = dst[e];
  const float alpha = ew[i] / (ssum[d * NHEAD + h] + 1e-16f);
  const float4* hp = (const float4*)(hfeat + (size_t)s * HD + h * DPH);
  const float4 h0 = hp[0], h1 = hp[1];
  float* op = out + (size_t)d * HD + h * DPH;
  atomicAdd(op + 0, alpha * h0.x); atomicAdd(op + 1, alpha * h0.y);
  atomicAdd(op + 2, alpha * h0.z); atomicAdd(op + 3, alpha * h0.w);
  atomicAdd(op + 4, alpha * h1.x); atomicAdd(op + 5, alpha * h1.y);
  atomicAdd(op + 6, alpha * h1.z); atomicAdd(op + 7, alpha * h1.w);
}

// ---------- epilogue: + bias, optional ELU ----------
__global__ void gat_epilogue(float* __restrict__ out, const float* __restrict__ bias,
                             int N64, int elu) {
  int i = blockIdx.x * blockDim.x + threadIdx.x;
  if (i >= N64) return;
  float v = out[i] + bias[i & (HD - 1)];
  if (elu) v = (v > 0.f) ? v : expm1f(v);
  out[i] = v;
}

extern "C" void kernel_launch(void* const* d_in, const int* in_sizes, int n_in,
                              void* d_out, int out_size, void* d_ws, size_t ws_size,
                              hipStream_t stream) {
  const float* x  = (const float*)d_in[0];
  const int*   ei = (const int*)d_in[1];
  const int N = in_sizes[0] / 128;   // 50000
  const int E = in_sizes[1] / 2;     // 1600000
  const int* src = ei;
  const int* dst = ei + E;

  // workspace layout
  char* ws = (char*)d_ws;
  size_t o = 0;
  float*    bufX = (float*)(ws + o);    o += (size_t)N * HD * sizeof(float);      // ping-pong features
  float*    hbuf = (float*)(ws + o);    o += (size_t)N * HD * sizeof(float);      // GEMM result h
  float*    es   = (float*)(ws + o);    o += (size_t)N * NHEAD * sizeof(float);
  float*    ed   = (float*)(ws + o);    o += (size_t)N * NHEAD * sizeof(float);
  unsigned* mx   = (unsigned*)(ws + o); o += (size_t)N * NHEAD * sizeof(unsigned);
  float*    ssum = (float*)(ws + o);    o += (size_t)N * NHEAD * sizeof(float);
  float*    ew   = (float*)(ws + o);    // E*NHEAD floats
  (void)ws_size; (void)n_in; (void)out_size;

  const int N8  = N * NHEAD;
  const int N64 = N * HD;
  const int EH  = E * NHEAD;

  for (int layer = 0; layer < 4; ++layer) {
    const float* W   = (const float*)d_in[2 + 3 * layer];
    const float* att = (const float*)d_in[3 + 3 * layer];
    const float* b   = (const float*)d_in[4 + 3 * layer];
    const float* xin = (layer == 0) ? x : bufX;
    const int    K   = (layer == 0) ? (in_sizes[2] / HD) : HD;  // 128 then 64
    const int    ksh = (K == 128) ? 7 : 6;
    float*       agg = (layer == 3) ? (float*)d_out : bufX;     // input dead after GEMM

    const int strips = (N + 15) / 16;                           // 16-row strips
    gat_gemm_wmma<<<(strips + 3) / 4, 128, 0, stream>>>(xin, W, hbuf, N, K, ksh);
    gat_node_logits<<<(N8 + 255) / 256, 256, 0, stream>>>(hbuf, att, es, ed, N8);
    gat_init<<<(N64 + 255) / 256, 256, 0, stream>>>(ssum, mx, agg, N8, N64);
    gat_edge_max<<<(EH + 255) / 256, 256, 0, stream>>>(src, dst, es, ed, ew, mx, EH);
    gat_edge_exp<<<(EH + 255) / 256, 256, 0, stream>>>(dst, ew, mx, ssum, EH);
    gat_edge_scatter<<<(EH + 255) / 256, 256, 0, stream>>>(src, dst, ew, ssum, hbuf, agg, EH);
    gat_epilogue<<<(N64 + 255) / 256, 256, 0, stream>>>(agg, b, N64, (layer < 3) ? 1 : 0);
  }
}